// Regularization_2035814498880
// MI455X (gfx1250) — compile-verified
//
#include <hip/hip_runtime.h>
#include <hip/hip_bf16.h>
#include <math.h>

typedef __attribute__((ext_vector_type(2))) float v2f;
typedef __attribute__((ext_vector_type(8))) float v8f;

#define THREADS 256

// Accumulator slots (doubles at start of d_ws)
// 0: edge  1: normal  2: laplacian  3: iter

// Block-wide sum of `partial` over 256 threads, added (as double) to *acc_slot.
// Wave 0 reduces the 16x16 LDS tile with 4 chained V_WMMA_F32_16X16X4_F32:
// with A == all-ones, D[m][n] = sum_k B[k][n], so the sum of ALL 256 D elements
// equals 16 * (sum of all staged partials) regardless of operand lane layout.
__device__ __forceinline__ void block_reduce_add(float partial, double* acc_slot) {
  __shared__ float lds[THREADS];
  lds[threadIdx.x] = partial;
  __syncthreads();
  if (threadIdx.x < 32) {
    const int lane = (int)threadIdx.x;
#if __has_builtin(__builtin_amdgcn_wmma_f32_16x16x4_f32)
    v2f ones;
    ones[0] = 1.0f;
    ones[1] = 1.0f;
    v8f d = {};
#pragma unroll
    for (int c = 0; c < 4; ++c) {
      v2f b;
      b[0] = lds[c * 64 + lane];
      b[1] = lds[c * 64 + 32 + lane];
      // (neg_a, A, neg_b, B, c_mod, C, reuse_a, reuse_b)
      d = __builtin_amdgcn_wmma_f32_16x16x4_f32(false, ones, false, b,
                                                (short)0, d, false, false);
    }
    float s = 0.0f;
#pragma unroll
    for (int i = 0; i < 8; ++i) s += d[i];
#pragma unroll
    for (int off = 16; off > 0; off >>= 1) s += __shfl_down(s, off, 32);
    if (lane == 0) atomicAdd(acc_slot, (double)(s * 0.0625f));
#else
    float s = lds[lane];
#pragma unroll
    for (int i = 1; i < 8; ++i) s += lds[lane + 32 * i];
#pragma unroll
    for (int off = 16; off > 0; off >>= 1) s += __shfl_down(s, off, 32);
    if (lane == 0) atomicAdd(acc_slot, (double)s);
#endif
  }
}

__global__ void zero_ws_kernel(float* ws, int n) {
  int stride = gridDim.x * blockDim.x;
  for (int i = blockIdx.x * blockDim.x + threadIdx.x; i < n; i += stride)
    ws[i] = 0.0f;
}

// loss_edge partial + scatter of degree / neighbor sums for the Laplacian.
__global__ void edge_kernel(const float* __restrict__ verts,
                            const int* __restrict__ edges, int E,
                            float* __restrict__ deg, float* __restrict__ nbr,
                            double* __restrict__ acc) {
  float partial = 0.0f;
  int stride = gridDim.x * blockDim.x;
  for (int e = blockIdx.x * blockDim.x + threadIdx.x; e < E; e += stride) {
    int a = edges[2 * e + 0];
    int b = edges[2 * e + 1];
    float ax = verts[3 * a + 0], ay = verts[3 * a + 1], az = verts[3 * a + 2];
    float bx = verts[3 * b + 0], by = verts[3 * b + 1], bz = verts[3 * b + 2];
    float dx = ax - bx, dy = ay - by, dz = az - bz;
    partial += dx * dx + dy * dy + dz * dz;
    atomicAdd(&deg[a], 1.0f);
    atomicAdd(&deg[b], 1.0f);
    atomicAdd(&nbr[3 * a + 0], bx);
    atomicAdd(&nbr[3 * a + 1], by);
    atomicAdd(&nbr[3 * a + 2], bz);
    atomicAdd(&nbr[3 * b + 0], ax);
    atomicAdd(&nbr[3 * b + 1], ay);
    atomicAdd(&nbr[3 * b + 2], az);
  }
  block_reduce_add(partial, acc + 0);
}

__global__ void normal_kernel(const float* __restrict__ verts,
                              const int* __restrict__ ie_v,
                              const int* __restrict__ ie_opp, int IE,
                              double* __restrict__ acc) {
  float partial = 0.0f;
  int stride = gridDim.x * blockDim.x;
  for (int e = blockIdx.x * blockDim.x + threadIdx.x; e < IE; e += stride) {
    int i0 = ie_v[2 * e + 0];
    int i1 = ie_v[2 * e + 1];
    int ia = ie_opp[2 * e + 0];
    int ib = ie_opp[2 * e + 1];
    float v0x = verts[3 * i0 + 0], v0y = verts[3 * i0 + 1], v0z = verts[3 * i0 + 2];
    float v1x = verts[3 * i1 + 0], v1y = verts[3 * i1 + 1], v1z = verts[3 * i1 + 2];
    float axp = verts[3 * ia + 0], ayp = verts[3 * ia + 1], azp = verts[3 * ia + 2];
    float bxp = verts[3 * ib + 0], byp = verts[3 * ib + 1], bzp = verts[3 * ib + 2];
    float ex = v1x - v0x, ey = v1y - v0y, ez = v1z - v0z;
    float pax = axp - v0x, pay = ayp - v0y, paz = azp - v0z;
    float pbx = bxp - v0x, pby = byp - v0y, pbz = bzp - v0z;
    // n0 = cross(e, pa)
    float n0x = ey * paz - ez * pay;
    float n0y = ez * pax - ex * paz;
    float n0z = ex * pay - ey * pax;
    // n1 = -cross(e, pb)
    float n1x = -(ey * pbz - ez * pby);
    float n1y = -(ez * pbx - ex * pbz);
    float n1z = -(ex * pby - ey * pbx);
    float dp = n0x * n1x + n0y * n1y + n0z * n1z;
    float l0 = sqrtf(n0x * n0x + n0y * n0y + n0z * n0z);
    float l1 = sqrtf(n1x * n1x + n1y * n1y + n1z * n1z);
    float cosv = dp / (fmaxf(l0, 1e-8f) * fmaxf(l1, 1e-8f));
    partial += 1.0f - cosv;
  }
  block_reduce_add(partial, acc + 1);
}

__global__ void lap_kernel(const float* __restrict__ verts,
                           const float* __restrict__ deg,
                           const float* __restrict__ nbr, int V,
                           double* __restrict__ acc) {
  float partial = 0.0f;
  int stride = gridDim.x * blockDim.x;
  for (int v = blockIdx.x * blockDim.x + threadIdx.x; v < V; v += stride) {
    float inv = 1.0f / fmaxf(deg[v], 1.0f);
    float lx = nbr[3 * v + 0] * inv - verts[3 * v + 0];
    float ly = nbr[3 * v + 1] * inv - verts[3 * v + 1];
    float lz = nbr[3 * v + 2] * inv - verts[3 * v + 2];
    partial += sqrtf(lx * lx + ly * ly + lz * lz);
  }
  block_reduce_add(partial, acc + 2);
}

__global__ void iter_kernel(const float* __restrict__ J,
                            const unsigned char* __restrict__ masks, int F,
                            double* __restrict__ acc) {
  float partial = 0.0f;
  int stride = gridDim.x * blockDim.x;
  for (int f = blockIdx.x * blockDim.x + threadIdx.x; f < F; f += stride) {
    const float* jf = J + 9 * f;
#pragma unroll
    for (int r = 0; r < 3; ++r) {
      // mask true  -> J row = iter_jacobians row, contributes (J - I)^2
      // mask false -> J row = I row, contributes 0
      if (masks[3 * f + r] != 0) {
#pragma unroll
        for (int c = 0; c < 3; ++c) {
          float t = jf[3 * r + c] - ((r == c) ? 1.0f : 0.0f);
          partial += t * t;
        }
      }
    }
  }
  block_reduce_add(partial, acc + 3);
}

__global__ void finalize_kernel(const double* __restrict__ acc,
                                float* __restrict__ out, int E, int IE, int V,
                                int F) {
  if (blockIdx.x == 0 && threadIdx.x == 0) {
    double loss_edge = acc[0] / (double)E;
    double loss_normal = acc[1] / (double)IE;
    double loss_lap = acc[2] / (double)V;
    double iter_loss = acc[3] / ((double)F * 9.0);
    double loss = (loss_normal + loss_lap) * 0.1 + iter_loss * 0.25;
    out[0] = (float)loss;
    out[1] = (float)loss_edge;
    out[2] = (float)loss_normal;
    out[3] = (float)loss_lap;
    out[4] = (float)iter_loss;
  }
}

extern "C" void kernel_launch(void* const* d_in, const int* in_sizes, int n_in,
                              void* d_out, int out_size, void* d_ws,
                              size_t ws_size, hipStream_t stream) {
  (void)n_in;
  (void)out_size;
  (void)ws_size;

  const float* verts = (const float*)d_in[0];
  const float* iterJ = (const float*)d_in[1];
  // d_in[2] residual_jacobians: only feeds residual_loss, which is never output
  const unsigned char* masks = (const unsigned char*)d_in[3];  // jax bool = u8
  const int* edges = (const int*)d_in[4];
  const int* ie_v = (const int*)d_in[5];
  const int* ie_opp = (const int*)d_in[6];
  float* out = (float*)d_out;

  const int V = in_sizes[0] / 3;
  const int F = in_sizes[1] / 9;
  const int E = in_sizes[4] / 2;
  const int IE = in_sizes[5] / 2;

  // Workspace layout: 8 doubles (accumulators) | deg[V] f32 | nbr[3V] f32
  double* acc = (double*)d_ws;
  float* deg = (float*)((char*)d_ws + 64);
  float* nbr = deg + V;

  const int zn = 16 + 4 * V;  // 8 doubles (=16 floats) + deg + nbr
  zero_ws_kernel<<<(zn + THREADS - 1) / THREADS, THREADS, 0, stream>>>(
      (float*)d_ws, zn);

  edge_kernel<<<(E + THREADS - 1) / THREADS, THREADS, 0, stream>>>(
      verts, edges, E, deg, nbr, acc);
  normal_kernel<<<(IE + THREADS - 1) / THREADS, THREADS, 0, stream>>>(
      verts, ie_v, ie_opp, IE, acc);
  lap_kernel<<<(V + THREADS - 1) / THREADS, THREADS, 0, stream>>>(
      verts, deg, nbr, V, acc);
  iter_kernel<<<(F + THREADS - 1) / THREADS, THREADS, 0, stream>>>(
      iterJ, masks, F, acc);
  finalize_kernel<<<1, 64, 0, stream>>>(acc, out, E, IE, V, F);
}